// Poisson_42150809043188
// MI455X (gfx1250) — compile-verified
//
#include <hip/hip_runtime.h>
#include <hip/hip_bf16.h>
#include <stdint.h>

// Poisson rate-coding spike encoder for MI455X (gfx1250).
// No matrix contraction exists in this op -> WMMA is inapplicable.
// CDNA5-specific paths used instead:
//   * global_load_async_to_lds_b128 + s_wait_asynccnt (async-to-LDS path)
//   * non-temporal b128 streaming stores for the 314 MB write-once output
//   * wave32-sized blocking (256 threads = 8 waves)

#define TW 100          // TIME_WINDOW
#define RATIO_F 1.0f    // RATIO

typedef __attribute__((ext_vector_type(4))) float v4f;

// ---------------- counter-based RNG (deterministic per pixel) ----------------
__device__ __forceinline__ uint32_t pcg32(uint64_t& st) {
  uint64_t old = st;
  st = old * 6364136223846793005ULL + 1442695040888963407ULL;
  uint32_t xs  = (uint32_t)(((old >> 18u) ^ old) >> 27u);
  uint32_t rot = (uint32_t)(old >> 59u);
  return (xs >> rot) | (xs << ((32u - rot) & 31u));
}

__device__ __forceinline__ float u01(uint64_t& st) {
  // (0, 1] so log() is finite
  return (float)((pcg32(st) >> 8) + 1u) * 0x1p-24f;
}

__device__ __forceinline__ float expdraw(uint64_t& st) {
  return -__logf(u01(st));
}

// Poisson(lam) via Exp(1) accumulation; capped at TW+1 since any interval
// larger than the window ends the pixel anyway (bounds work for huge lam).
__device__ __forceinline__ int poisson_trunc(float lam, uint64_t& st) {
  float s = expdraw(st);
  int k = 0;
  while (s <= lam) {
    if (++k > TW + 1) break;
    s += expdraw(st);
  }
  return k;
}

// Build the 100-bit spike mask for one pixel.
__device__ __forceinline__ void encode_pixel(float x, uint64_t seed, uint32_t m[4]) {
  m[0] = m[1] = m[2] = m[3] = 0u;
  float v = x * RATIO_F;
  if (v == 0.0f) return;                       // zero pixels: no spikes
  float lam = 1.0f / v;
  uint64_t st = seed;
  (void)pcg32(st);                             // warm up state
  int tcum = 0;
#pragma unroll 1
  for (int i = 0; i < TW; ++i) {               // at most TW spikes (intervals >= 1)
    int s  = poisson_trunc(lam, st);
    int iv = (s < 1) ? 1 : s;                  // intervals += (intervals == 0)
    tcum += iv;
    if (tcum > TW) break;                      // times >= T+1 are dropped
    int b = tcum - 1;                          // spikes[1:] shift
    m[b >> 5] |= (1u << (b & 31));
  }
}

__global__ __launch_bounds__(256)
void poisson_encode_kernel(const float* __restrict__ img,
                           float* __restrict__ out, int n) {
  __shared__ __align__(16) float tile[256 * 4];
  const int tid = threadIdx.x;
  const long long gid  = (long long)blockIdx.x * blockDim.x + tid;
  const long long base = gid * 4;

  // ---- stage 16 B/thread of input through LDS via async DMA (ASYNCcnt) ----
  uint32_t lds_addr = (uint32_t)(uintptr_t)(&tile[tid * 4]);
  if (base + 3 < (long long)n) {
    const float* g = img + base;
    asm volatile("global_load_async_to_lds_b128 %0, %1, off"
                 :: "v"(lds_addr), "v"(g)
                 : "memory");
  } else {
    for (int i = 0; i < 4; ++i)
      tile[tid * 4 + i] = (base + i < (long long)n) ? img[base + i] : 0.0f;
  }
  asm volatile("s_wait_asynccnt 0x0" ::: "memory");

  if (base >= (long long)n) return;

  // ---- per-pixel spike masks (registers) ----
  uint32_t m[4][4];
#pragma unroll
  for (int p = 0; p < 4; ++p) {
    float x = tile[tid * 4 + p];
    uint64_t seed = (0x9E3779B97F4A7C15ULL * (uint64_t)(base + p + 1)) ^ 0x2A2A2A2AULL;
    encode_pixel(x, seed, m[p]);
  }

  // ---- stream the [T, N] output: coalesced non-temporal b128 stores ----
  if (((n & 3) == 0) && (base + 3 < (long long)n)) {
    const long long rowStride = (long long)(n >> 2);       // in v4f units
    v4f* p = (v4f*)out + gid;                              // running row pointer
#pragma unroll 1
    for (int w = 0; w < 4; ++w) {                          // mask word (t>>5 const)
      const uint32_t w0 = m[0][w], w1 = m[1][w], w2 = m[2][w], w3 = m[3][w];
      const int nbits = (w < 3) ? 32 : (TW - 96);          // 32,32,32,4
#pragma unroll 4
      for (int b = 0; b < nbits; ++b) {
        v4f v;
        v.x = (float)((w0 >> b) & 1u);
        v.y = (float)((w1 >> b) & 1u);
        v.z = (float)((w2 >> b) & 1u);
        v.w = (float)((w3 >> b) & 1u);
        __builtin_nontemporal_store(v, p);
        p += rowStride;
      }
    }
  } else {
#pragma unroll 1
    for (int t = 0; t < TW; ++t)
      for (int p4 = 0; p4 < 4; ++p4)
        if (base + p4 < (long long)n)
          out[(long long)t * n + base + p4] =
              (float)((m[p4][t >> 5] >> (t & 31)) & 1u);
  }
}

extern "C" void kernel_launch(void* const* d_in, const int* in_sizes, int n_in,
                              void* d_out, int out_size, void* d_ws, size_t ws_size,
                              hipStream_t stream) {
  (void)n_in; (void)d_ws; (void)ws_size; (void)out_size;
  const float* img = (const float*)d_in[0];
  float* out = (float*)d_out;
  int n = in_sizes[0];                       // 64*3*64*64 = 786432 pixels
  int threads_needed = (n + 3) / 4;          // 4 pixels per thread
  int block = 256;                           // 8 wave32 per block
  int grid = (threads_needed + block - 1) / block;
  poisson_encode_kernel<<<grid, block, 0, stream>>>(img, out, n);
}